// FP_Layer_8813272891484
// MI455X (gfx1250) — compile-verified
//
#include <hip/hip_runtime.h>
#include <hip/hip_bf16.h>

typedef __attribute__((ext_vector_type(2))) float v2f;
typedef __attribute__((ext_vector_type(8))) float v8f;

#define B_    8
#define N1_   8192
#define N2_   2048
#define C1_   128
#define C2_   256
#define INCH_ 384
#define OC_   256
#define EPS_D_  1e-8f
#define EPS_BN_ 1e-5f

// ---------------------------------------------------------------------------
// Kernel 1: brute-force 3-NN + inverse-distance interpolation + concat
// grid = (N1/256, B), block = 256. xyz2 (24KB) staged in LDS; the j-loop
// reads it as wave-uniform broadcasts (conflict-free).
// ---------------------------------------------------------------------------
__global__ __launch_bounds__(256) void knn_interp_concat(
    const float* __restrict__ xyz1, const float* __restrict__ xyz2,
    const float* __restrict__ feats1, const float* __restrict__ feats2,
    float* __restrict__ catbuf)
{
    __shared__ float sx[N2_ * 3];
    const int b   = blockIdx.y;
    const int tid = threadIdx.x;

    for (int e = tid; e < N2_ * 3; e += 256)
        sx[e] = xyz2[b * N2_ * 3 + e];
    __syncthreads();

    const int n = blockIdx.x * 256 + tid;
    const float px = xyz1[(b * N1_ + n) * 3 + 0];
    const float py = xyz1[(b * N1_ + n) * 3 + 1];
    const float pz = xyz1[(b * N1_ + n) * 3 + 2];

    float bd0 = 3.4e38f, bd1 = 3.4e38f, bd2 = 3.4e38f;
    int   bi0 = 0, bi1 = 0, bi2 = 0;

    for (int j = 0; j < N2_; ++j) {
        float dx = px - sx[j * 3 + 0];
        float dy = py - sx[j * 3 + 1];
        float dz = pz - sx[j * 3 + 2];
        float d2 = dx * dx + dy * dy + dz * dz;
        if (d2 < bd2) {
            bd2 = d2; bi2 = j;
            if (bd2 < bd1) { float t = bd1; bd1 = bd2; bd2 = t; int ti = bi1; bi1 = bi2; bi2 = ti; }
            if (bd1 < bd0) { float t = bd0; bd0 = bd1; bd1 = t; int ti = bi0; bi0 = bi1; bi1 = ti; }
        }
    }

    float d0 = fmaxf(sqrtf(fmaxf(bd0, 0.f)), EPS_D_);
    float d1 = fmaxf(sqrtf(fmaxf(bd1, 0.f)), EPS_D_);
    float d2 = fmaxf(sqrtf(fmaxf(bd2, 0.f)), EPS_D_);
    float w0 = 1.f / d0, w1 = 1.f / d1, w2 = 1.f / d2;
    float ws = 1.f / (w0 + w1 + w2);
    w0 *= ws; w1 *= ws; w2 *= ws;

    const float* f2 = feats2 + (size_t)b * C2_ * N2_;
    float* cb = catbuf + (size_t)b * INCH_ * N1_;

#pragma unroll 4
    for (int c = 0; c < C2_; ++c) {
        const float* row = f2 + c * N2_;
        cb[c * N1_ + n] = w0 * row[bi0] + w1 * row[bi1] + w2 * row[bi2];
    }

    const float* f1 = feats1 + (size_t)b * C1_ * N1_;
#pragma unroll 4
    for (int c = 0; c < C1_; ++c)
        cb[(C2_ + c) * N1_ + n] = f1[c * N1_ + n];
}

// ---------------------------------------------------------------------------
// Kernel 2: Y[b][o][n] = sum_k W[o][k] * X[b][k][n] + bias[o]
// via V_WMMA_F32_16X16X4_F32. grid = (N1/32, B), block = 256 (8 waves).
// A 32-column activation panel [K][32] is staged transposed into LDS
// (stride 390 floats -> conflict-free writes, 8B-aligned b64 reads).
// Each wave computes a 2x2 grid of 16x16 tiles (2 M-tiles x 2 N-tiles):
// every A fragment feeds 2 WMMAs and every B fragment feeds 2 WMMAs.
// ---------------------------------------------------------------------------
template <int K>
__global__ __launch_bounds__(256) void gemm_wmma(
    const float* __restrict__ W, const float* __restrict__ bias,
    const float* __restrict__ X, float* __restrict__ Y)
{
    constexpr int LDW = 390;             // padded stride (floats): even, 390%64==6
    __shared__ float Xs[32 * LDW];

    const int b     = blockIdx.y;
    const int ntile = blockIdx.x * 32;
    const int tid   = threadIdx.x;

    const float* Xb = X + (size_t)b * K * N1_;
    for (int e = tid; e < K * 32; e += 256) {
        int k  = e >> 5;
        int nn = e & 31;
        Xs[nn * LDW + k] = Xb[k * N1_ + ntile + nn];   // transpose into LDS
    }
    __syncthreads();

    const int wave = tid >> 5;
    const int lane = tid & 31;
    const int half = lane >> 4;          // 0: K pair {0,1}, 1: K pair {2,3}
    const int nl   = lane & 15;          // B/C/D column within a 16-wide tile
    const int ma   = lane & 15;          // A row

    const int m0 = (wave * 2 + 0) * 16;  // OC=256 -> 16 M-tiles, 2 per wave
    const int m1 = (wave * 2 + 1) * 16;

    v8f acc00 = {}; v8f acc01 = {};      // [mTile][nTile]
    v8f acc10 = {}; v8f acc11 = {};

    const float* lds0 = &Xs[nl * LDW];
    const float* lds1 = &Xs[(16 + nl) * LDW];

    for (int k0 = 0; k0 < K; k0 += 4) {
        const int kk = k0 + 2 * half;
        v2f b0 = *(const v2f*)&lds0[kk];
        v2f b1 = *(const v2f*)&lds1[kk];

        const float* wp0 = W + (m0 + ma) * K + kk;
        const float* wp1 = W + (m1 + ma) * K + kk;
        v2f a0; a0.x = wp0[0]; a0.y = wp0[1];
        v2f a1; a1.x = wp1[0]; a1.y = wp1[1];

        acc00 = __builtin_amdgcn_wmma_f32_16x16x4_f32(false, a0, false, b0,
                                                      (short)0, acc00, false, false);
        acc01 = __builtin_amdgcn_wmma_f32_16x16x4_f32(false, a0, false, b1,
                                                      (short)0, acc01, false, false);
        acc10 = __builtin_amdgcn_wmma_f32_16x16x4_f32(false, a1, false, b0,
                                                      (short)0, acc10, false, false);
        acc11 = __builtin_amdgcn_wmma_f32_16x16x4_f32(false, a1, false, b1,
                                                      (short)0, acc11, false, false);
    }

    float* Yb = Y + (size_t)b * OC_ * N1_;
#pragma unroll
    for (int r = 0; r < 8; ++r) {
        int mm = r + 8 * half;           // C/D layout: VGPR r -> M = r (+8 hi half)
        float bias0 = bias[m0 + mm];
        float bias1 = bias[m1 + mm];
        Yb[(m0 + mm) * N1_ + ntile +      nl] = acc00[r] + bias0;
        Yb[(m0 + mm) * N1_ + ntile + 16 + nl] = acc01[r] + bias0;
        Yb[(m1 + mm) * N1_ + ntile +      nl] = acc10[r] + bias1;
        Yb[(m1 + mm) * N1_ + ntile + 16 + nl] = acc11[r] + bias1;
    }
}

// ---------------------------------------------------------------------------
// Kernel 3: per-channel mean / rstd over (B, N). One block per channel,
// deterministic LDS tree reduction (no atomics -> no zero-init pass).
// ---------------------------------------------------------------------------
__global__ __launch_bounds__(256) void bn_stats(
    const float* __restrict__ Y, float* __restrict__ stats)
{
    __shared__ float ssum[256];
    __shared__ float ssq[256];
    const int o = blockIdx.x;

    float s = 0.f, q = 0.f;
    for (int idx = threadIdx.x; idx < B_ * N1_; idx += 256) {
        int bb = idx >> 13;              // / N1_
        int nn = idx & (N1_ - 1);
        float v = Y[((size_t)bb * OC_ + o) * N1_ + nn];
        s += v; q += v * v;
    }
    ssum[threadIdx.x] = s;
    ssq[threadIdx.x]  = q;
    __syncthreads();
    for (int off = 128; off > 0; off >>= 1) {
        if (threadIdx.x < off) {
            ssum[threadIdx.x] += ssum[threadIdx.x + off];
            ssq[threadIdx.x]  += ssq[threadIdx.x + off];
        }
        __syncthreads();
    }
    if (threadIdx.x == 0) {
        const float inv = 1.f / (float)(B_ * N1_);
        float mean = ssum[0] * inv;
        float var  = ssq[0] * inv - mean * mean;
        stats[o]       = mean;
        stats[OC_ + o] = rsqrtf(var + EPS_BN_);
    }
}

// ---------------------------------------------------------------------------
// Kernel 4: y = relu(g * (x - mean) * rstd + beta), elementwise.
// ---------------------------------------------------------------------------
__global__ __launch_bounds__(256) void bn_relu(
    const float* __restrict__ Yin, float* __restrict__ Yout,
    const float* __restrict__ stats,
    const float* __restrict__ g, const float* __restrict__ beta)
{
    size_t i = (size_t)blockIdx.x * 256 + threadIdx.x;
    int o = (int)((i >> 13) & (OC_ - 1));    // layout [b][o][n], N1 = 2^13
    float m = stats[o];
    float r = stats[OC_ + o];
    float v = g[o] * ((Yin[i] - m) * r) + beta[o];
    Yout[i] = fmaxf(v, 0.f);
}

// ---------------------------------------------------------------------------
extern "C" void kernel_launch(void* const* d_in, const int* in_sizes, int n_in,
                              void* d_out, int out_size, void* d_ws, size_t ws_size,
                              hipStream_t stream)
{
    const float* xyz1   = (const float*)d_in[0];
    const float* xyz2   = (const float*)d_in[1];
    const float* feats1 = (const float*)d_in[2];
    const float* feats2 = (const float*)d_in[3];
    const float* w1     = (const float*)d_in[4];
    const float* b1     = (const float*)d_in[5];
    const float* g1     = (const float*)d_in[6];
    const float* beta1  = (const float*)d_in[7];
    const float* w2     = (const float*)d_in[8];
    const float* b2     = (const float*)d_in[9];
    const float* g2     = (const float*)d_in[10];
    const float* beta2  = (const float*)d_in[11];

    float* out   = (float*)d_out;
    float* cat0  = (float*)d_ws;                                   // (B,384,N1)
    float* stats = (float*)((char*)d_ws + (size_t)B_ * INCH_ * N1_ * sizeof(float));
    float* y2    = cat0;                                           // reuse cat region

    // 1) 3-NN interpolation + concat -> ws
    knn_interp_concat<<<dim3(N1_ / 256, B_), 256, 0, stream>>>(
        xyz1, xyz2, feats1, feats2, cat0);

    // 2) conv1: y1 = w1 . cat + b1 -> d_out (scratch use)
    gemm_wmma<INCH_><<<dim3(N1_ / 32, B_), 256, 0, stream>>>(w1, b1, cat0, out);

    // 3) BN1 stats + relu in place -> h in d_out
    bn_stats<<<OC_, 256, 0, stream>>>(out, stats);
    bn_relu<<<(B_ * OC_ * N1_) / 256, 256, 0, stream>>>(out, out, stats, g1, beta1);

    // 4) conv2: y2 = w2 . h + b2 -> ws (overwrites cat, no longer needed)
    gemm_wmma<OC_><<<dim3(N1_ / 32, B_), 256, 0, stream>>>(w2, b2, out, y2);

    // 5) BN2 stats + relu -> final d_out
    bn_stats<<<OC_, 256, 0, stream>>>(y2, stats);
    bn_relu<<<(B_ * OC_ * N1_) / 256, 256, 0, stream>>>(y2, out, stats, g2, beta2);
}